// FeaturesMap_35107062677845
// MI455X (gfx1250) — compile-verified
//
#include <hip/hip_runtime.h>
#include <hip/hip_bf16.h>

// ---------------------------------------------------------------------------
// FeaturesMap: bbox -> (transpose?) -> centered crop/pad shift -> scatter.
// Implemented as scatter of point-ids into a tiny index map, then a fully
// coalesced gather into the output. Bandwidth-bound (~160 MB total traffic).
// ---------------------------------------------------------------------------

namespace {
constexpr int kB = 16;
constexpr int kC = 512;
constexpr int kN = 2048;
constexpr int kH = 70;
constexpr int kW = 40;
constexpr int kPlane = kH * kW;        // 2800
constexpr int kBig = 1000000;
constexpr int kThreads = 256;
constexpr int kCPB = 8;                // channels per gather block
}

typedef float  f4v __attribute__((ext_vector_type(4)));
typedef int    i4v __attribute__((ext_vector_type(4)));

// Types matching the async-LDS builtin signature (from hipcc diagnostic):
// param0 = global (AS1) pointer to 16-byte int vector, param1 = LDS (AS3).
typedef int fm_v4i __attribute__((vector_size(4 * sizeof(int))));
typedef __attribute__((address_space(1))) fm_v4i fm_gv4;
typedef __attribute__((address_space(3))) fm_v4i fm_sv4;

#if defined(__has_builtin)
#if __has_builtin(__builtin_amdgcn_global_load_async_to_lds_b128) && \
    __has_builtin(__builtin_amdgcn_s_wait_asynccnt)
#define FM_USE_ASYNC_LDS 1
#endif
#endif

// ---------------------------------------------------------------------------
// Kernel A: one block per sample. Reduce bbox, derive transform, build the
// index map plane (n or -1) in workspace. atomicMax resolves the (rare)
// duplicate pixels produced by JAX's negative-index wrap-then-drop semantics.
// ---------------------------------------------------------------------------
__global__ __launch_bounds__(kThreads)
void fm_build_index(const int* __restrict__ ys, const int* __restrict__ xs,
                    int* __restrict__ idxmap) {
  const int b   = blockIdx.x;
  const int tid = threadIdx.x;
  const int* __restrict__ y = ys + b * kN;
  const int* __restrict__ x = xs + b * kN;

  int miny = kBig, maxy = -kBig, minx = kBig, maxx = -kBig;
  for (int n = tid; n < kN; n += kThreads) {
    const int yv = y[n];
    const int xv = x[n];
    const bool v = yv > -1;
    miny = min(miny, v ? yv : kBig);
    maxy = max(maxy, v ? yv : -kBig);
    minx = min(minx, v ? xv : kBig);
    maxx = max(maxx, v ? xv : -kBig);
  }

  __shared__ int s0[kThreads], s1[kThreads], s2[kThreads], s3[kThreads];
  s0[tid] = miny; s1[tid] = maxy; s2[tid] = minx; s3[tid] = maxx;
  __syncthreads();
  for (int off = kThreads / 2; off > 0; off >>= 1) {
    if (tid < off) {
      s0[tid] = min(s0[tid], s0[tid + off]);
      s1[tid] = max(s1[tid], s1[tid + off]);
      s2[tid] = min(s2[tid], s2[tid + off]);
      s3[tid] = max(s3[tid], s3[tid + off]);
    }
    __syncthreads();
  }
  miny = s0[0]; maxy = s1[0]; minx = s2[0]; maxx = s3[0];

  const int h     = maxy - miny + 1;
  const int w     = maxx - minx + 1;
  const bool tr   = w > h;
  const int H2    = tr ? w : h;
  const int W2    = tr ? h : w;
  const int hd    = H2 - kH;
  const int wd    = W2 - kW;
  const int sy    = (hd > 0) ? -((hd + 1) >> 1) : ((1 - hd) >> 1);
  const int sx    = (wd > 0) ? -((wd + 1) >> 1) : ((1 - wd) >> 1);

  int* __restrict__ plane = idxmap + b * kPlane;
  for (int p = tid; p < kPlane; p += kThreads) plane[p] = -1;
  __syncthreads();  // -1 fill visible before the scatter below

  for (int n = tid; n < kN; n += kThreads) {
    const int yv = y[n];
    const int xv = x[n];
    if (yv > -1) {
      const int yy = tr ? (xv - minx) : (yv - miny);
      const int xx = tr ? (yv - miny) : (xv - minx);
      int oy = yy + sy;
      int ox = xx + sx;
      // JAX normalizes negative indices once (idx += dim), then drops OOB.
      if (oy < 0) oy += kH;
      if (ox < 0) ox += kW;
      if (oy >= 0 && oy < kH && ox >= 0 && ox < kW)
        atomicMax(&plane[oy * kW + ox], n);
    }
  }
}

// ---------------------------------------------------------------------------
// Kernel B: gather. Block = (sample b, 8-channel slab). Index plane staged in
// LDS via gfx1250 async global->LDS DMA (ASYNCcnt) when available. Output is
// written with coalesced 128-bit nontemporal stores (write-once stream).
// ---------------------------------------------------------------------------
__global__ __launch_bounds__(kThreads)
void fm_gather(const float* __restrict__ features,
               const float* __restrict__ backend,
               const int* __restrict__ idxmap,
               float* __restrict__ out) {
  const int b   = blockIdx.x;
  const int c0  = blockIdx.y * kCPB;
  const int tid = threadIdx.x;

  __shared__ __align__(16) int sidx[kPlane];
  const int* __restrict__ plane = idxmap + b * kPlane;

#ifdef FM_USE_ASYNC_LDS
  for (int q = tid; q < kPlane / 4; q += kThreads) {
    __builtin_amdgcn_global_load_async_to_lds_b128(
        (fm_gv4*)(plane + 4 * q),
        (fm_sv4*)(sidx + 4 * q),
        /*offset=*/0, /*cpol=*/0);
  }
  __builtin_amdgcn_s_wait_asynccnt(0);
  __syncthreads();
#else
  for (int p = tid; p < kPlane; p += kThreads) sidx[p] = plane[p];
  __syncthreads();
#endif

  const float* __restrict__ fb = features + (size_t)b * kC * kN;
  float* __restrict__ ob = out + ((size_t)b * kC + c0) * kPlane;

#pragma unroll
  for (int ci = 0; ci < kCPB; ++ci) {
    const int c = c0 + ci;
    const float* __restrict__ fc = fb + (size_t)c * kN;
    const float bk = backend[c];
    f4v* __restrict__ oc4 = (f4v*)(ob + (size_t)ci * kPlane);

    for (int q = tid; q < kPlane / 4; q += kThreads) {
      const i4v n4 = ((const i4v*)sidx)[q];
      f4v v;
      v.x = (n4.x >= 0) ? fc[n4.x] : bk;
      v.y = (n4.y >= 0) ? fc[n4.y] : bk;
      v.z = (n4.z >= 0) ? fc[n4.z] : bk;
      v.w = (n4.w >= 0) ? fc[n4.w] : bk;
      __builtin_nontemporal_store(v, oc4 + q);  // write-once stream -> NT
    }
  }
}

// ---------------------------------------------------------------------------
extern "C" void kernel_launch(void* const* d_in, const int* in_sizes, int n_in,
                              void* d_out, int out_size, void* d_ws,
                              size_t ws_size, hipStream_t stream) {
  (void)in_sizes; (void)n_in; (void)out_size; (void)ws_size;
  const float* features = (const float*)d_in[0];   // [B, C, N] f32
  const float* backend  = (const float*)d_in[1];   // [C, 1, 1] f32
  const int*   ys       = (const int*)d_in[2];     // [B, N] i32
  const int*   xs       = (const int*)d_in[3];     // [B, N] i32
  float*       out      = (float*)d_out;           // [B, C, 70, 40] f32
  int*         idxmap   = (int*)d_ws;              // [B, 2800] i32 scratch

  fm_build_index<<<kB, kThreads, 0, stream>>>(ys, xs, idxmap);
  fm_gather<<<dim3(kB, kC / kCPB), kThreads, 0, stream>>>(features, backend,
                                                          idxmap, out);
}